// MemoryQueue_8942121910790
// MI455X (gfx1250) — compile-verified
//
#include <hip/hip_runtime.h>

typedef __attribute__((ext_vector_type(2))) float v2f;
typedef __attribute__((ext_vector_type(8))) float v8f;

#define DDIM        256        // feature dim
#define TINV        20.0f      // 1/T, T = 0.05
#define ROWS_PER_WG 16
#define WAVES       8
#define COLS_ITER   (WAVES * 32)   // 256 columns per iteration (2 tiles/wave)
#define QCHUNKS     16
#define LDS_STRIDE  260        // 256 + 4 pad -> lanes land on distinct banks

// ---------- workspace layout ----------
// [0      .. 8192 )  anchor-id bitmap, 2048 u32 (65536 bits)
// [8192   .. 40960)  per-row best key, 4096 u64
// [40960  .. 106496) column mask, 65536 bytes
#define WS_ROWBEST_OFF 8192
#define WS_COLMASK_OFF 40960

__device__ __forceinline__ unsigned map_f32(float v) {
    unsigned b = __float_as_uint(v);
    return (b & 0x80000000u) ? ~b : (b | 0x80000000u);   // monotone f32 -> u32
}

__device__ __forceinline__ unsigned long long shfl_xor_u64(unsigned long long v, int m) {
    unsigned lo = (unsigned)v, hi = (unsigned)(v >> 32);
    lo = __shfl_xor(lo, m, 32);
    hi = __shfl_xor(hi, m, 32);
    return ((unsigned long long)hi << 32) | lo;
}

// ---------- 1) init bitmap + row bests ----------
__global__ void mq_init(unsigned* __restrict__ bm, unsigned long long* __restrict__ rowbest,
                        int nbm, int nrows) {
    int t = blockIdx.x * blockDim.x + threadIdx.x;
    if (t < nbm)   bm[t] = 0u;
    if (t < nrows) rowbest[t] = 0ull;
}

// ---------- 2) scatter anchor ids ----------
__global__ void mq_scatter(const int* __restrict__ id_anchors, unsigned* __restrict__ bm, int n) {
    int t = blockIdx.x * blockDim.x + threadIdx.x;
    if (t < n) {
        unsigned id = (unsigned)id_anchors[t] & 0xFFFFu;
        atomicOr(&bm[id >> 5], 1u << (id & 31u));
    }
}

// ---------- 3) column mask byte per q ----------
__global__ void mq_colmask(const int* __restrict__ mem_id, const unsigned* __restrict__ bm,
                           unsigned char* __restrict__ colmask, int q) {
    int t = blockIdx.x * blockDim.x + threadIdx.x;
    if (t < q) {
        unsigned id = (unsigned)mem_id[t] & 0xFFFFu;
        colmask[t] = (unsigned char)((bm[id >> 5] >> (id & 31u)) & 1u);
    }
}

// ---------- 4) WMMA GEMM + fused masked max/argmax ----------
// Each wave: 2 independent 16x16 tiles x 2 split-K accumulators
// -> 4 independent v_wmma_f32_16x16x4_f32 chains to fill the matrix pipe.
__global__ void __launch_bounds__(256)
mq_gemm(const float* __restrict__ anchors, const float* __restrict__ mem_feat,
        const unsigned char* __restrict__ colmask,
        unsigned long long* __restrict__ rowbest, int Q) {
    __shared__ float Alds[ROWS_PER_WG * LDS_STRIDE];

    const int tid  = threadIdx.x;
    const int wave = tid >> 5;
    const int lane = tid & 31;
    const int m    = lane & 15;       // A row / B col within tile
    const int half = lane >> 4;       // K-pair select
    const int rowBase = blockIdx.y * ROWS_PER_WG;
    const int qChunk  = Q / QCHUNKS;                 // 4096
    const int qBase   = blockIdx.x * qChunk;

    // Stage A tile (16 x 256 f32) into LDS, padded row stride.
    for (int i = tid; i < ROWS_PER_WG * (DDIM / 4); i += 256) {
        int r  = i >> 6;              // / (DDIM/4)
        int k4 = (i & 63) << 2;
        float4 v = *(const float4*)(anchors + (size_t)(rowBase + r) * DDIM + k4);
        float* dst = &Alds[r * LDS_STRIDE + k4];
        dst[0] = v.x; dst[1] = v.y; dst[2] = v.z; dst[3] = v.w;
    }
    __syncthreads();

    unsigned long long best[8];
#pragma unroll
    for (int r = 0; r < 8; ++r) best[r] = 0ull;

    const float* aptr = &Alds[m * LDS_STRIDE + 2 * half];
    const int iters = qChunk / COLS_ITER;            // 16

    for (int it = 0; it < iters; ++it) {
        const int q0 = qBase + it * COLS_ITER + wave * 32;   // tile0 cols q0..q0+15
        const int q1 = q0 + 16;                              // tile1 cols q1..q1+15
        const float* bptr0 = mem_feat + (size_t)(q0 + m) * DDIM + 2 * half;
        const float* bptr1 = mem_feat + (size_t)(q1 + m) * DDIM + 2 * half;

        v8f acc00 = {0.f,0.f,0.f,0.f,0.f,0.f,0.f,0.f};  // tile0, even K-steps
        v8f acc01 = {0.f,0.f,0.f,0.f,0.f,0.f,0.f,0.f};  // tile0, odd  K-steps
        v8f acc10 = {0.f,0.f,0.f,0.f,0.f,0.f,0.f,0.f};  // tile1, even K-steps
        v8f acc11 = {0.f,0.f,0.f,0.f,0.f,0.f,0.f,0.f};  // tile1, odd  K-steps
#pragma unroll 8
        for (int kk = 0; kk < DDIM; kk += 8) {
            v2f a0  = *(const v2f*)(aptr + kk);        // ds_load (2addr-merged)
            v2f a1  = *(const v2f*)(aptr + kk + 4);
            v2f b00 = *(const v2f*)(bptr0 + kk);       // global_load_b64, L2-hot
            v2f b01 = *(const v2f*)(bptr0 + kk + 4);
            v2f b10 = *(const v2f*)(bptr1 + kk);
            v2f b11 = *(const v2f*)(bptr1 + kk + 4);
            acc00 = __builtin_amdgcn_wmma_f32_16x16x4_f32(
                false, a0, false, b00, (short)0, acc00, false, false);
            acc10 = __builtin_amdgcn_wmma_f32_16x16x4_f32(
                false, a0, false, b10, (short)0, acc10, false, false);
            acc01 = __builtin_amdgcn_wmma_f32_16x16x4_f32(
                false, a1, false, b01, (short)0, acc01, false, false);
            acc11 = __builtin_amdgcn_wmma_f32_16x16x4_f32(
                false, a1, false, b11, (short)0, acc11, false, false);
        }

        // fused column mask (-1/T), scale by 1/T, running packed argmax
        const int col0 = q0 + m;
        const int col1 = q1 + m;
        const unsigned msk0 = colmask[col0];
        const unsigned msk1 = colmask[col1];
        const unsigned idxKey0 = 0xFFFFFFFFu - (unsigned)col0;  // lower col wins ties
        const unsigned idxKey1 = 0xFFFFFFFFu - (unsigned)col1;
#pragma unroll
        for (int r = 0; r < 8; ++r) {
            float v0 = msk0 ? -TINV : (acc00[r] + acc01[r]) * TINV;
            float v1 = msk1 ? -TINV : (acc10[r] + acc11[r]) * TINV;
            unsigned long long k0 =
                ((unsigned long long)map_f32(v0) << 32) | idxKey0;
            unsigned long long k1 =
                ((unsigned long long)map_f32(v1) << 32) | idxKey1;
            if (k1 > k0) k0 = k1;
            if (k0 > best[r]) best[r] = k0;
        }
    }

    // reduce across the 16 columns held by each half-wave (xor 1,2,4,8)
#pragma unroll
    for (int r = 0; r < 8; ++r) {
        unsigned long long k = best[r];
#pragma unroll
        for (int s = 1; s < 16; s <<= 1) {
            unsigned long long o = shfl_xor_u64(k, s);
            if (o > k) k = o;
        }
        best[r] = k;
    }
    if (m == 0) {  // lanes 0 (rows 0..7) and 16 (rows 8..15)
        const int row = rowBase + half * 8;
#pragma unroll
        for (int r = 0; r < 8; ++r)
            atomicMax(&rowbest[row + r], best[r]);
    }
}

// ---------- 5) unpack value + gather nearest row ----------
__global__ void __launch_bounds__(256)
mq_finalize(const float* __restrict__ mem_feat,
            const unsigned long long* __restrict__ rowbest,
            float* __restrict__ out, int B) {
    const int b = blockIdx.x;
    const unsigned long long p = rowbest[b];
    const unsigned idx = 0xFFFFFFFFu - (unsigned)(p & 0xFFFFFFFFu);
    const unsigned mu  = (unsigned)(p >> 32);
    const unsigned bits = (mu & 0x80000000u) ? (mu ^ 0x80000000u) : ~mu;
    if (threadIdx.x == 0) out[b] = __uint_as_float(bits);
    out[(size_t)B + (size_t)b * DDIM + threadIdx.x] =
        mem_feat[(size_t)idx * DDIM + threadIdx.x];
}

extern "C" void kernel_launch(void* const* d_in, const int* in_sizes, int n_in,
                              void* d_out, int out_size, void* d_ws, size_t ws_size,
                              hipStream_t stream) {
    const float* anchors    = (const float*)d_in[0];   // [B, 256] f32
    const float* mem_feat   = (const float*)d_in[1];   // [Q, 256] f32
    const int*   mem_id     = (const int*)  d_in[2];   // [Q]
    const int*   id_anchors = (const int*)  d_in[3];   // [B]
    float* out = (float*)d_out;                        // values[B] ++ nearest[B*256]

    const int B = in_sizes[3];     // 4096
    const int Q = in_sizes[2];     // 65536

    unsigned*            bm      = (unsigned*)d_ws;
    unsigned long long*  rowbest = (unsigned long long*)((char*)d_ws + WS_ROWBEST_OFF);
    unsigned char*       colmask = (unsigned char*)d_ws + WS_COLMASK_OFF;

    mq_init<<<(B + 255) / 256, 256, 0, stream>>>(bm, rowbest, Q / 32, B);
    mq_scatter<<<(B + 255) / 256, 256, 0, stream>>>(id_anchors, bm, B);
    mq_colmask<<<(Q + 255) / 256, 256, 0, stream>>>(mem_id, bm, colmask, Q);

    dim3 grid(QCHUNKS, B / ROWS_PER_WG);               // 16 x 256 = 4096 WGs
    mq_gemm<<<grid, 256, 0, stream>>>(anchors, mem_feat, colmask, rowbest, Q);

    mq_finalize<<<B, DDIM, 0, stream>>>(mem_feat, rowbest, out, B);
}